// DifferentiablePermutation_38963943309495
// MI455X (gfx1250) — compile-verified
//
#include <hip/hip_runtime.h>
#include <math.h>

typedef __attribute__((ext_vector_type(2))) float v2f;
typedef __attribute__((ext_vector_type(4))) float v4f;
typedef __attribute__((ext_vector_type(8))) float v8f;

#define SIZE 4096
#define EPSV 1e-8f

// ---------------------------------------------------------------------------
// Pass 1: row softmax.  One block (256 threads) per row; the 16 KB row is
// held in registers between the max pass and the exp/sum pass.
// ---------------------------------------------------------------------------
__global__ void softmax_rows_kernel(const float* __restrict__ logits,
                                    float* __restrict__ S) {
  __shared__ float red[256];
  const int row = blockIdx.x;
  const float* in = logits + (size_t)row * SIZE;
  float* out = S + (size_t)row * SIZE;

  float vals[16];
  float m = -INFINITY;
#pragma unroll
  for (int i = 0; i < 16; ++i) {
    vals[i] = in[threadIdx.x + i * 256];
    m = fmaxf(m, vals[i]);
  }
  red[threadIdx.x] = m;
  __syncthreads();
  for (int s = 128; s > 0; s >>= 1) {
    if (threadIdx.x < s)
      red[threadIdx.x] = fmaxf(red[threadIdx.x], red[threadIdx.x + s]);
    __syncthreads();
  }
  m = red[0];
  __syncthreads();

  float acc = 0.0f;
#pragma unroll
  for (int i = 0; i < 16; ++i) {
    vals[i] = expf(vals[i] - m);
    acc += vals[i];
  }
  red[threadIdx.x] = acc;
  __syncthreads();
  for (int s = 128; s > 0; s >>= 1) {
    if (threadIdx.x < s) red[threadIdx.x] += red[threadIdx.x + s];
    __syncthreads();
  }
  const float inv = 1.0f / red[0];
#pragma unroll
  for (int i = 0; i < 16; ++i) out[threadIdx.x + i * 256] = vals[i] * inv;
}

// ---------------------------------------------------------------------------
// Init Sinkhorn scaling vectors r = c = 1.
// ---------------------------------------------------------------------------
__global__ void init_rc_kernel(float* __restrict__ r, float* __restrict__ c) {
  const int i = blockIdx.x * blockDim.x + threadIdx.x;
  if (i < SIZE) {
    r[i] = 1.0f;
    c[i] = 1.0f;
  }
}

// ---------------------------------------------------------------------------
// Row pass: u = S*c (16-row strip per block), then r_i <- r_i/(r_i*u_i+eps).
// WMMA f32 16x16x4: A = 16x4 tile of S, B = c-chunk broadcast across N.
// A layout (ISA 7.12.2, 32-bit A 16x4): lane m (0..15), half h = lane>>4:
//   VGPR0 = A[m][2h], VGPR1 = A[m][2h+1] -> one contiguous 8B load per lane.
// B uses the same K-slot order, so the contraction pairs S[.,k] with c[k]
// regardless of the exact hardware slot permutation.
// 8 waves split K (512 cols each, 128 WMMAs); partials reduced via LDS.
// ---------------------------------------------------------------------------
__global__ void sinkhorn_row_kernel(const float* __restrict__ S,
                                    float* __restrict__ r,
                                    const float* __restrict__ c) {
  __shared__ float part[8 * 16];
  const int lane = threadIdx.x & 31;
  const int wave = threadIdx.x >> 5;
  const int half = lane >> 4;
  const int m = lane & 15;
  const int r0 = blockIdx.x * 16;

  const float* arow = S + (size_t)(r0 + m) * SIZE + 2 * half;
  const int kb = wave * (SIZE / 8);
  const int ke = kb + (SIZE / 8);

  v8f acc = {};
#pragma unroll 8
  for (int k0 = kb; k0 < ke; k0 += 4) {
    v2f a = *(const v2f*)(arow + k0);  // S[r0+m][k0+2h .. k0+2h+1]
    v2f b;
    b.x = c[k0 + 2 * half];
    b.y = c[k0 + 2 * half + 1];
    acc = __builtin_amdgcn_wmma_f32_16x16x4_f32(false, a, false, b,
                                                (short)0, acc, false, false);
  }
  // D layout: lanes with N==0 (m==0) hold rows r0 + 8*half + v in acc[v].
  if (m == 0) {
#pragma unroll
    for (int v = 0; v < 8; ++v) part[wave * 16 + half * 8 + v] = acc[v];
  }
  __syncthreads();
  if (threadIdx.x < 16) {
    float u = 0.0f;
#pragma unroll
    for (int w = 0; w < 8; ++w) u += part[w * 16 + threadIdx.x];
    const int row = r0 + threadIdx.x;
    const float rv = r[row];
    r[row] = rv / (rv * u + EPSV);
  }
}

// ---------------------------------------------------------------------------
// Col pass: v = S^T*r (16-column strip per block), then
// c_j <- c_j/(c_j*v_j+eps).  A[m][k] = S[k0+k][j0+m]: each K-slot load is a
// 16-float contiguous row segment (64B coalesced per half-wave).
// ---------------------------------------------------------------------------
__global__ void sinkhorn_col_kernel(const float* __restrict__ S,
                                    const float* __restrict__ r,
                                    float* __restrict__ c) {
  __shared__ float part[8 * 16];
  const int lane = threadIdx.x & 31;
  const int wave = threadIdx.x >> 5;
  const int half = lane >> 4;
  const int m = lane & 15;
  const int j0 = blockIdx.x * 16;

  const int kb = wave * (SIZE / 8);
  const int ke = kb + (SIZE / 8);

  v8f acc = {};
#pragma unroll 8
  for (int k0 = kb; k0 < ke; k0 += 4) {
    const size_t base = (size_t)(k0 + 2 * half) * SIZE + j0 + m;
    v2f a;
    a.x = S[base];         // S[k0+2h  ][j0+m]
    a.y = S[base + SIZE];  // S[k0+2h+1][j0+m]
    v2f b;
    b.x = r[k0 + 2 * half];
    b.y = r[k0 + 2 * half + 1];
    acc = __builtin_amdgcn_wmma_f32_16x16x4_f32(false, a, false, b,
                                                (short)0, acc, false, false);
  }
  if (m == 0) {
#pragma unroll
    for (int v = 0; v < 8; ++v) part[wave * 16 + half * 8 + v] = acc[v];
  }
  __syncthreads();
  if (threadIdx.x < 16) {
    float u = 0.0f;
#pragma unroll
    for (int w = 0; w < 8; ++w) u += part[w * 16 + threadIdx.x];
    const int col = j0 + threadIdx.x;
    const float cv = c[col];
    c[col] = cv / (cv * u + EPSV);
  }
}

// ---------------------------------------------------------------------------
// Final: out_ij = r_i * S_ij * c_j, in place over d_out (float4 vectorized).
// ---------------------------------------------------------------------------
__global__ void scale_out_kernel(float* __restrict__ M,
                                 const float* __restrict__ r,
                                 const float* __restrict__ c) {
  const size_t tid = (size_t)blockIdx.x * blockDim.x + threadIdx.x;
  const size_t idx = tid * 4;
  const int row = (int)(idx >> 12);  // / 4096
  const int col = (int)(idx & 4095);
  const float rv = r[row];
  v4f x = *(v4f*)(M + idx);
  x.x *= rv * c[col + 0];
  x.y *= rv * c[col + 1];
  x.z *= rv * c[col + 2];
  x.w *= rv * c[col + 3];
  *(v4f*)(M + idx) = x;
}

// ---------------------------------------------------------------------------
extern "C" void kernel_launch(void* const* d_in, const int* in_sizes, int n_in,
                              void* d_out, int out_size, void* d_ws,
                              size_t ws_size, hipStream_t stream) {
  (void)in_sizes; (void)n_in; (void)out_size; (void)ws_size;
  const float* logits = (const float*)d_in[0];
  float* S = (float*)d_out;  // S lives in d_out; scaled in place at the end
  float* r = (float*)d_ws;   // 4096 floats
  float* c = r + SIZE;       // 4096 floats

  softmax_rows_kernel<<<SIZE, 256, 0, stream>>>(logits, S);
  init_rc_kernel<<<SIZE / 256, 256, 0, stream>>>(r, c);

  for (int it = 0; it < 50; ++it) {
    sinkhorn_row_kernel<<<SIZE / 16, 256, 0, stream>>>(S, r, c);
    sinkhorn_col_kernel<<<SIZE / 16, 256, 0, stream>>>(S, r, c);
  }

  scale_out_kernel<<<(SIZE * (SIZE / 4)) / 256, 256, 0, stream>>>(S, r, c);
}